// Attention_21775484191401
// MI455X (gfx1250) — compile-verified
//
#include <hip/hip_runtime.h>

// Problem dims (fixed by the reference).
constexpr int NB = 32;
constexpr int NT = 4096;
constexpr int NH = 1024;

// Fusion geometry.
constexpr int S     = 16;                   // T-slabs per batch -> NB*S = 512 workgroups
constexpr int SLAB  = NT / S;               // 256 rows per slab
constexpr int WAVES = 8;                    // 256-thread block = 8 wave32
constexpr int TILES = SLAB / (WAVES * 16);  // 2 tiles of 16 rows per wave

typedef __attribute__((ext_vector_type(2))) float v2f;
typedef __attribute__((ext_vector_type(4))) float v4f;
typedef __attribute__((ext_vector_type(8))) float v8f;

// ---------------------------------------------------------------------------
// Kernel 1: single-pass flash attention-pooling over a T-slab.
// Per 16-row tile (per wave):
//   Phase A: scores via V_WMMA_F32_16X16X4_F32
//     A (16x4 f32): lane m=lane&15 holds A[m, koff..koff+1], koff=2*(lane>>4)
//     B (4x16 f32): prev broadcast to all N columns (same K mapping)
//     -> every D column equals the 16 scores. REGULAR-temporal loads so the
//        64KB tile stays hot in WGP$/L2 for Phase B.
//   Online softmax update of (m, l); rescale C.
//   Phase B: C[h] += w_j * hist[row_j, h], last-use NT loads (cache hits).
// Cross-wave combine: ordered LDS reduction (deterministic, no FP atomics).
// ---------------------------------------------------------------------------
__global__ __launch_bounds__(256) void
fused_flash(const float* __restrict__ hist, const float* __restrict__ prev,
            float* __restrict__ scores,     // raw scores -> attn slot of d_out
            float* __restrict__ part_ctx,   // [NB*S, NH]
            float* __restrict__ part_ml)    // [NB*S, 2] = (M, L)
{
  __shared__ float s_prev[NH];              // 4 KB
  __shared__ float s_wctx[WAVES][NH];       // 32 KB per-wave context stripes
  __shared__ float s_sc[WAVES][16];         // per-wave tile scores
  __shared__ float s_ml[WAVES][2];          // per-wave (m, l)

  const int b    = blockIdx.x / S;
  const int slab = blockIdx.x % S;
  const int tid  = threadIdx.x;

  ((v4f*)s_prev)[tid] = ((const v4f*)(prev + (size_t)b * NH))[tid];
  __syncthreads();

  const int wave = tid >> 5;
  const int lane = tid & 31;
  const int mrow = lane & 15;
  const int koff = (lane >> 4) << 1;        // 0 or 2

  float m = -3.402823466e38f;
  float l = 0.0f;
  float C[32];                              // lane covers h = 4*lane + 128*q + p
#pragma unroll
  for (int r = 0; r < 32; ++r) C[r] = 0.0f;

  for (int tile = 0; tile < TILES; ++tile) {
    const int t0 = slab * SLAB + (tile * WAVES + wave) * 16;

    // ---------------- Phase A: 16 scores via WMMA (cache-resident loads) ---
    const float* aRow = hist + ((size_t)b * NT + (t0 + mrow)) * NH + koff;
    const float* bCol = s_prev + koff;
    v8f c = {0.f, 0.f, 0.f, 0.f, 0.f, 0.f, 0.f, 0.f};
#pragma unroll 8
    for (int k = 0; k < NH; k += 4) {
      v2f a = *(const v2f*)(aRow + k);      // RT: keep lines for Phase B
      v2f bb;
      bb.x = bCol[k];
      bb.y = bCol[k + 1];
      c = __builtin_amdgcn_wmma_f32_16x16x4_f32(false, a, false, bb, (short)0,
                                                c, false, false);
    }
    // D: VGPR r -> M=r (lanes 0-15) / M=r+8 (lanes 16-31); all N equal.
    float* sout = scores + (size_t)b * NT + t0;
    if (lane == 0) {
#pragma unroll
      for (int j = 0; j < 8; ++j) { sout[j] = c[j]; s_sc[wave][j] = c[j]; }
    } else if (lane == 16) {
#pragma unroll
      for (int j = 0; j < 8; ++j) { sout[8 + j] = c[j]; s_sc[wave][8 + j] = c[j]; }
    }
    __syncthreads();                        // uniform; publishes s_sc

    // ---------------- Online softmax update -------------------------------
    float w[16];
    float tmax = -3.402823466e38f;
#pragma unroll
    for (int j = 0; j < 16; ++j) { w[j] = s_sc[wave][j]; tmax = fmaxf(tmax, w[j]); }
    const float newM  = fmaxf(m, tmax);
    const float scale = expf(m - newM);     // first tile: exp(-huge) = 0
    float sum = 0.f;
#pragma unroll
    for (int j = 0; j < 16; ++j) { w[j] = expf(w[j] - newM); sum += w[j]; }
    l = l * scale + sum;
    m = newM;
#pragma unroll
    for (int r = 0; r < 32; ++r) C[r] *= scale;

    // ---------------- Phase B: rank-16 update, last-use streaming re-read --
    const float* rowp = hist + ((size_t)b * NT + t0) * NH + 4 * lane;
    for (int j = 0; j < 16; ++j) {
      const float* rp = rowp + (size_t)j * NH;
      const float wj = w[j];
#pragma unroll
      for (int q = 0; q < 8; ++q) {
        v4f hv = __builtin_nontemporal_load((const v4f*)(rp + 128 * q));
        C[q * 4 + 0] = fmaf(wj, hv.x, C[q * 4 + 0]);
        C[q * 4 + 1] = fmaf(wj, hv.y, C[q * 4 + 1]);
        C[q * 4 + 2] = fmaf(wj, hv.z, C[q * 4 + 2]);
        C[q * 4 + 3] = fmaf(wj, hv.w, C[q * 4 + 3]);
      }
    }
    __syncthreads();                        // uniform; s_sc reusable next tile
  }

  // ---------------- Deterministic cross-wave combine -----------------------
  if (lane == 0) { s_ml[wave][0] = m; s_ml[wave][1] = l; }
  float M = -3.402823466e38f;
  // Publish this wave's C (scaled to block max) into its own LDS stripe.
  __syncthreads();
#pragma unroll
  for (int i = 0; i < WAVES; ++i) M = fmaxf(M, s_ml[i][0]);
  float L = 0.f;
#pragma unroll
  for (int i = 0; i < WAVES; ++i) L += s_ml[i][1] * expf(s_ml[i][0] - M);
  const float myscale = expf(m - M);
#pragma unroll
  for (int r = 0; r < 32; ++r)
    s_wctx[wave][4 * lane + 128 * (r >> 2) + (r & 3)] = C[r] * myscale;
  __syncthreads();

  // Fixed-order sum over the 8 wave stripes: deterministic.
  v4f acc = {0.f, 0.f, 0.f, 0.f};
#pragma unroll
  for (int i = 0; i < WAVES; ++i) acc += ((const v4f*)s_wctx[i])[tid];
  ((v4f*)(part_ctx + (size_t)blockIdx.x * NH))[tid] = acc;
  if (tid == 0) {
    part_ml[2 * blockIdx.x]     = M;
    part_ml[2 * blockIdx.x + 1] = L;
  }
}

// ---------------------------------------------------------------------------
// Kernel 2: flash combine over the S slab partials of one batch.
// context[b,h] = (1/L) * sum_i exp(M_i - M) * C_i[h];  also stores (M, 1/L).
// ---------------------------------------------------------------------------
__global__ __launch_bounds__(256) void
combine_ctx(const float* __restrict__ part_ctx, const float* __restrict__ part_ml,
            float* __restrict__ ctx, float* __restrict__ ml_final)
{
  const int b   = blockIdx.x;
  const int tid = threadIdx.x;

  float ms[S], ls[S];
  float M = -3.402823466e38f;
#pragma unroll
  for (int i = 0; i < S; ++i) {
    ms[i] = part_ml[2 * (b * S + i)];
    ls[i] = part_ml[2 * (b * S + i) + 1];
    M = fmaxf(M, ms[i]);
  }
  float L = 0.f, sc[S];
#pragma unroll
  for (int i = 0; i < S; ++i) { sc[i] = expf(ms[i] - M); L += ls[i] * sc[i]; }
  const float invL = 1.0f / L;

  v4f acc = {0.f, 0.f, 0.f, 0.f};
#pragma unroll
  for (int i = 0; i < S; ++i) {
    v4f p = ((const v4f*)(part_ctx + (size_t)(b * S + i) * NH))[tid];
    acc += p * sc[i];
  }
  ((v4f*)(ctx + (size_t)b * NH))[tid] = acc * invL;

  if (tid == 0) { ml_final[2 * b] = M; ml_final[2 * b + 1] = invL; }
}

// ---------------------------------------------------------------------------
// Kernel 3: attn[b,t] = exp(s_t - M) * (1/L), in place on the attn slot.
// ---------------------------------------------------------------------------
__global__ __launch_bounds__(256) void
finalize_attn(float* __restrict__ attn, const float* __restrict__ ml_final)
{
  const int b   = blockIdx.x;
  const int tid = threadIdx.x;
  const float M    = ml_final[2 * b];
  const float invL = ml_final[2 * b + 1];
  float* s = attn + (size_t)b * NT;
#pragma unroll
  for (int i = 0; i < 16; ++i) {
    const int t = tid + i * 256;
    s[t] = expf(s[t] - M) * invL;
  }
}

// ---------------------------------------------------------------------------
// Harness entry. Output layout: context [B*H] then attn_weights [B*T].
// hist_h is streamed from HBM exactly once (Phase-B re-reads hit cache);
// workspace = slab partials (~2.1 MB), fully rewritten every call.
// ---------------------------------------------------------------------------
extern "C" void kernel_launch(void* const* d_in, const int* in_sizes, int n_in,
                              void* d_out, int out_size, void* d_ws,
                              size_t ws_size, hipStream_t stream) {
  const float* hist = (const float*)d_in[0];     // [B, T, H] fp32
  const float* prev = (const float*)d_in[1];     // [B, H, 1] fp32
  float* out  = (float*)d_out;
  float* ctx  = out;                             // B*H
  float* attn = out + (size_t)NB * NH;           // B*T (raw scores -> attn)

  float* ws       = (float*)d_ws;
  float* part_ctx = ws;                          // NB*S*NH floats
  float* part_ml  = part_ctx + (size_t)NB * S * NH;  // NB*S*2
  float* ml_final = part_ml + NB * S * 2;        // NB*2

  fused_flash<<<NB * S, 256, 0, stream>>>(hist, prev, attn, part_ctx, part_ml);
  combine_ctx<<<NB, 256, 0, stream>>>(part_ctx, part_ml, ctx, ml_final);
  finalize_attn<<<NB, 256, 0, stream>>>(attn, ml_final);
}